// CIF_Model_85521388798689
// MI455X (gfx1250) — compile-verified
//
#include <hip/hip_runtime.h>

// CIF (Continuous Integrate-and-Fire), B=32 T=2048 H=512 L=256.
// out[b] = W[b] @ hidden[b] with W a narrow monotone band; computed with
// V_WMMA_F32_16X16X4_F32 over the band only (~1.1 GFLOP vs 17.2 dense),
// leaving the kernel purely HBM-bound (~145 MB => ~6.3 us @ 23.3 TB/s).
//
// Phase 1 (cif_scan): parallel prefix-sum scan (alpha < 1 => fire count has
// closed form c_t = max(0, ceil(prefix_t - 0.95))). Emits one packed 16-byte
// record per timestep {w0, w1, slot} plus per-slot band bounds tFirst/tLast.
// Phase 2 (cif_wmma): per (batch, 16-label tile) block, 8 waves x 4 N-tiles,
// K-loop in chunks of 8 timesteps (2 WMMA groups). Meta records are
// software-pipelined one iteration ahead so the A-fragment build never
// stalls on its own load (removes the s_wait_loadcnt 0x0 seen in round 3).

#define THRESH 0.95f

typedef float v2f __attribute__((ext_vector_type(2)));
typedef float v4f __attribute__((ext_vector_type(4)));
typedef float v8f __attribute__((ext_vector_type(8)));

// ---------------------------------------------------------------------------
// Phase 1: parallel CIF scan.  One block (256 threads) per batch row.
// Assumes T == 2048 (8 elements/thread) and L <= 256.
// meta[b*T+t] = { w0, w1, bitcast(s0), 0 }  (w1 targets min(s0+1, L-1))
// ---------------------------------------------------------------------------
__global__ __launch_bounds__(256) void cif_scan_kernel(
    const float* __restrict__ alphas,   // [B,T]
    v4f*   __restrict__ meta,           // [B,T] packed per-timestep record
    int*   __restrict__ tFirst,         // [B,L]
    int*   __restrict__ tLast,          // [B,L]
    int T, int L)
{
  const int b   = blockIdx.x;
  const int tid = threadIdx.x;
  const int EPT = 8;                    // T / 256

  __shared__ float sc[256];
  __shared__ int ldsFirst[256];
  __shared__ int ldsLast[256];

  for (int l = tid; l < L; l += 256) { ldsFirst[l] = T; ldsLast[l] = -1; }

  // Serial prefix within the thread's 8 elements.
  const float* arow = alphas + (size_t)b * T;
  float a[8], p[8];
  float run = 0.f;
#pragma unroll
  for (int j = 0; j < 8; ++j) {
    a[j] = arow[tid * EPT + j];
    run += a[j];
    p[j] = run;
  }
  sc[tid] = run;
  __syncthreads();
  // Hillis-Steele inclusive block scan over thread totals.
#pragma unroll
  for (int off = 1; off < 256; off <<= 1) {
    float v = (tid >= off) ? sc[tid - off] : 0.f;
    __syncthreads();
    sc[tid] += v;
    __syncthreads();
  }
  const float excl   = sc[tid] - run;   // prefix before this thread's chunk
  const float totalP = sc[255];
  const int totalFires = max(0, (int)ceilf(totalP - THRESH));

  int   lastC      = max(0, (int)ceilf(excl - THRESH));  // c_{t-1}
  float prevPrefix = excl;
  const size_t base = (size_t)b * T + (size_t)tid * EPT;
#pragma unroll
  for (int j = 0; j < 8; ++j) {
    const int   t    = tid * EPT + j;
    const float pref = excl + p[j];
    const int   c    = max(0, (int)ceilf(pref - THRESH));
    const bool fired = c > lastC;
    int S0; float W0, W1;
    if (fired) {
      // dist = 1 - integrate_before = 1 - (prefix_{t-1} - c_{t-1})
      const float dist = 1.f - (prevPrefix - (float)lastC);
      S0 = min(c - 1, L - 1);
      W0 = dist;                                   // always emitted (this fire)
      W1 = (c < totalFires) ? (a[j] - dist) : 0.f; // rem -> next fire's slot
      const int S1 = min(c, L - 1);
      atomicMin(&ldsFirst[S1], t);
      atomicMax(&ldsLast[S1], t);
    } else {
      S0 = min(c, L - 1);
      W0 = (c < totalFires) ? a[j] : 0.f;          // trailing tail never emitted
      W1 = 0.f;
    }
    atomicMin(&ldsFirst[S0], t);
    atomicMax(&ldsLast[S0], t);
    v4f m;
    m.x = W0; m.y = W1; m.z = __int_as_float(S0); m.w = 0.f;
    meta[base + j] = m;
    lastC = c;
    prevPrefix = pref;
  }
  __syncthreads();
  for (int l = tid; l < L; l += 256) {
    tFirst[(size_t)b * L + l] = ldsFirst[l];
    tLast [(size_t)b * L + l] = ldsLast[l];
  }
}

// Build one A-fragment element from a packed meta record (pure cndmask).
static __device__ __forceinline__ float aval(v4f m, int row, int Lm1, bool valid) {
  const int S0 = __float_as_int(m.z);
  const int S1 = min(S0 + 1, Lm1);
  float av = ((S0 == row) ? m.x : 0.f) + ((S1 == row) ? m.y : 0.f);
  return valid ? av : 0.f;
}

// ---------------------------------------------------------------------------
// Phase 2: banded WMMA matmul.  grid = (L/16, B), block = 256 (8 wave32).
// Wave w owns columns [64w, 64w+64) = 4 N-tiles; assumes H == 512.
// VGPR layouts per CDNA5 ISA 7.12.2:
//   A 16x4 f32 : lane&15 = M row; element j holds K = 2*(lane>>4) + j
//   B 4x16 f32 : lane&15 = N col; element j holds K = 2*(lane>>4) + j
//   C/D 16x16  : element r -> M = r + 8*(lane>>4), N = lane&15
// ---------------------------------------------------------------------------
__global__ __launch_bounds__(256) void cif_wmma_kernel(
    const float* __restrict__ hidden,   // [B,T,H]
    const v4f*   __restrict__ meta,     // [B,T]
    const int*   __restrict__ tFirst,
    const int*   __restrict__ tLast,
    float*       __restrict__ out,      // [B,L,H]
    int T, int H, int L)
{
  const int b  = blockIdx.y;
  const int l0 = blockIdx.x * 16;

  __shared__ int shTS, shTE;
  if (threadIdx.x == 0) {
    int ts = T, te = -1;
    for (int i = 0; i < 16; ++i) {
      ts = min(ts, tFirst[(size_t)b * L + l0 + i]);
      te = max(te, tLast [(size_t)b * L + l0 + i]);
    }
    shTS = ts; shTE = te;
  }
  __syncthreads();
  const int ts = shTS, te = shTE;       // uniform across the block

  const int wave = threadIdx.x >> 5;
  const int lane = threadIdx.x & 31;
  const int half = lane >> 4;
  const int l16  = lane & 15;
  const int n0   = wave * 64;
  const int row  = l0 + l16;            // A-matrix row owned by this lane
  const int kb   = 2 * half;            // this lane's K base within a sub-chunk

  v8f acc0 = {}, acc1 = {}, acc2 = {}, acc3 = {};
  const float* hb = hidden + (size_t)b * T * H;    // 64-bit base, 32-bit offsets
  const v4f*   mb = meta   + (size_t)b * T;
  const int nc    = n0 + l16;
  const int Lm1   = L - 1;
  const int pfMax = (T - 1) * H + nc;              // clamped prefetch offset

  if (ts <= te) {                       // uniform guard (EXEC stays all-ones)
    // Prologue: meta records for the first K=8 chunk (indices in [0, te]).
    v4f m00 = mb[min(ts     + kb,     te)];
    v4f m01 = mb[min(ts     + kb + 1, te)];
    v4f m10 = mb[min(ts + 4 + kb,     te)];
    v4f m11 = mb[min(ts + 4 + kb + 1, te)];

    for (int tc = ts; tc <= te; tc += 8) {
      // ---- A fragments + B offsets for the CURRENT chunk (registers only).
      const int t00 = tc + kb,     t01 = t00 + 1;
      const int t10 = tc + 4 + kb, t11 = t10 + 1;
      v2f A0, A1;
      A0[0] = aval(m00, row, Lm1, t00 <= te);
      A0[1] = aval(m01, row, Lm1, t01 <= te);
      A1[0] = aval(m10, row, Lm1, t10 <= te);
      A1[1] = aval(m11, row, Lm1, t11 <= te);
      const int o00 = min(t00, te) * H + nc;
      const int o01 = min(t01, te) * H + nc;
      const int o10 = min(t10, te) * H + nc;
      const int o11 = min(t11, te) * H + nc;

      // ---- Pipeline: issue NEXT chunk's meta loads; latency hides under
      //      the B-load clause + 8 WMMAs below.
      const int tn = tc + 8;
      m00 = mb[min(tn     + kb,     te)];
      m01 = mb[min(tn     + kb + 1, te)];
      m10 = mb[min(tn + 4 + kb,     te)];
      m11 = mb[min(tn + 4 + kb + 1, te)];

      // One clamped prefetch per iteration, two chunks ahead.
      __builtin_prefetch(hb + min(o00 + 16 * H, pfMax), 0, 1);

      // ---- Sub-chunk 0: B loads + 4 WMMAs.
      v2f B0, B1, B2, B3;
      B0[0] = hb[o00 +  0]; B0[1] = hb[o01 +  0];
      B1[0] = hb[o00 + 16]; B1[1] = hb[o01 + 16];
      B2[0] = hb[o00 + 32]; B2[1] = hb[o01 + 32];
      B3[0] = hb[o00 + 48]; B3[1] = hb[o01 + 48];
      acc0 = __builtin_amdgcn_wmma_f32_16x16x4_f32(false, A0, false, B0, (short)0, acc0, false, false);
      acc1 = __builtin_amdgcn_wmma_f32_16x16x4_f32(false, A0, false, B1, (short)0, acc1, false, false);
      acc2 = __builtin_amdgcn_wmma_f32_16x16x4_f32(false, A0, false, B2, (short)0, acc2, false, false);
      acc3 = __builtin_amdgcn_wmma_f32_16x16x4_f32(false, A0, false, B3, (short)0, acc3, false, false);

      // ---- Sub-chunk 1: B loads + 4 WMMAs.
      v2f C0, C1, C2, C3;
      C0[0] = hb[o10 +  0]; C0[1] = hb[o11 +  0];
      C1[0] = hb[o10 + 16]; C1[1] = hb[o11 + 16];
      C2[0] = hb[o10 + 32]; C2[1] = hb[o11 + 32];
      C3[0] = hb[o10 + 48]; C3[1] = hb[o11 + 48];
      acc0 = __builtin_amdgcn_wmma_f32_16x16x4_f32(false, A1, false, C0, (short)0, acc0, false, false);
      acc1 = __builtin_amdgcn_wmma_f32_16x16x4_f32(false, A1, false, C1, (short)0, acc1, false, false);
      acc2 = __builtin_amdgcn_wmma_f32_16x16x4_f32(false, A1, false, C2, (short)0, acc2, false, false);
      acc3 = __builtin_amdgcn_wmma_f32_16x16x4_f32(false, A1, false, C3, (short)0, acc3, false, false);
    }
  }

  // Store D (covers every (l,n) exactly once; empty bands store zeros).
  float* ob = out + ((size_t)b * L + l0) * H + n0 + l16;   // one 64-bit addr
#pragma unroll
  for (int r = 0; r < 8; ++r) {
    const int mo = (r + 8 * half) * H;              // 32-bit row offset
    ob[mo +  0] = acc0[r];
    ob[mo + 16] = acc1[r];
    ob[mo + 32] = acc2[r];
    ob[mo + 48] = acc3[r];
  }
}

// ---------------------------------------------------------------------------
extern "C" void kernel_launch(void* const* d_in, const int* in_sizes, int n_in,
                              void* d_out, int out_size, void* d_ws, size_t ws_size,
                              hipStream_t stream) {
  const float* hidden = (const float*)d_in[0];   // [B,T,H] f32
  const float* alphas = (const float*)d_in[1];   // [B,T]   f32
  (void)n_in;

  const int B  = 32;                     // reference batch size
  const int BT = in_sizes[1];            // B*T
  const int T  = BT / B;                 // 2048
  const int H  = in_sizes[0] / BT;       // 512
  const int L  = (int)((size_t)out_size / ((size_t)B * H));  // 256

  char* ws = (char*)d_ws;
  v4f* metag = (v4f*)ws;    ws += (size_t)BT * sizeof(v4f);   // 16B-aligned first
  int* tF    = (int*)ws;    ws += (size_t)B * L * sizeof(int);
  int* tL    = (int*)ws;
  (void)ws_size;

  cif_scan_kernel<<<B, 256, 0, stream>>>(alphas, metag, tF, tL, T, L);

  dim3 grid(L / 16, B);
  cif_wmma_kernel<<<grid, 256, 0, stream>>>(hidden, metag, tF, tL,
                                            (float*)d_out, T, H, L);
}